// GNN_5961414607307
// MI455X (gfx1250) — compile-verified
//
#include <hip/hip_runtime.h>

typedef __attribute__((ext_vector_type(2))) float v2f;
typedef __attribute__((ext_vector_type(8))) float v8f;

#define DH 128
#define LDST 130          // padded LDS stride (8B-aligned float2, low bank conflicts)
#define NEG_SLOPE 0.2f

// ---------------- degree / normalization ----------------
__global__ void k_init_deg(float* deg, int N) {
  int i = blockIdx.x * blockDim.x + threadIdx.x;
  if (i < N) deg[i] = 1.0f;                       // self loop
}

__global__ void k_count_deg(const long long* __restrict__ ei, float* deg, int E) {
  int e = blockIdx.x * blockDim.x + threadIdx.x;
  if (e < E) atomicAdd(&deg[(int)ei[(size_t)E + e]], 1.0f);   // col = ei[E..2E)
}

__global__ void k_rsqrt(float* d, int N) {
  int i = blockIdx.x * blockDim.x + threadIdx.x;
  if (i < N) d[i] = rsqrtf(d[i]);
}

// ---------------- f32 WMMA GEMM: T[N,128] = X[N,128] @ W[128,128] ----------------
// one wave = one 16-row tile x all 128 cols (8 x v8f accumulators)
__global__ void k_gemm_wmma(const float* __restrict__ X, const float* __restrict__ W,
                            float* __restrict__ T, int N, int ntiles) {
  __shared__ float Wl[DH * LDST];                 // W transposed: Wl[c*LDST + r] = W[r][c]
  for (int idx = threadIdx.x; idx < DH * DH; idx += blockDim.x) {
    int r = idx >> 7, c = idx & 127;
    Wl[c * LDST + r] = W[idx];
  }
  __syncthreads();

  int wave = threadIdx.x >> 5;
  int lane = threadIdx.x & 31;
  int tile = blockIdx.x * (blockDim.x >> 5) + wave;
  if (tile >= ntiles) return;                     // uniform per wave -> EXEC all-1s inside

  int row0 = tile << 4;
  int m  = lane & 15;                             // row (A) / col (B,D) within 16
  int kh = (lane >> 4) << 1;                      // K sub-offset: lanes 16-31 handle K+2,K+3

  int ra = row0 + m; if (ra > N - 1) ra = N - 1;
  const float* xr = X + (size_t)ra * DH;

  v8f acc[8] = {};
  for (int k = 0; k < DH; k += 4) {
    v2f a;
    a.x = xr[k + kh];
    a.y = xr[k + kh + 1];
#pragma unroll
    for (int t = 0; t < 8; ++t) {
      const float* wp = &Wl[(t * 16 + m) * LDST + k + kh];
      v2f b; b.x = wp[0]; b.y = wp[1];
      acc[t] = __builtin_amdgcn_wmma_f32_16x16x4_f32(
          false, a, false, b, (short)0, acc[t], false, false);
    }
  }

  int rbase = row0 + ((lane >> 4) << 3);          // lanes 16-31 hold rows M+8
#pragma unroll
  for (int t = 0; t < 8; ++t) {
#pragma unroll
    for (int v = 0; v < 8; ++v) {
      int rr = rbase + v;
      if (rr < N) T[(size_t)rr * DH + t * 16 + m] = acc[t][v];
    }
  }
}

// ---------------- out = residual + T*dis^2 + b ----------------
__global__ void k_init_out(float* __restrict__ out, const float* __restrict__ prev,
                           const float* __restrict__ t, const float* __restrict__ dis,
                           const float* __restrict__ bias, int N) {
  int idx = blockIdx.x * blockDim.x + threadIdx.x;  // over N*32 float4 slots
  if (idx >= N * 32) return;
  int n = idx >> 5;
  int d4 = (idx & 31) << 2;
  float s = dis[n]; float s2 = s * s;
  size_t base = (size_t)n * DH + d4;
#pragma unroll
  for (int j = 0; j < 4; ++j) {
    float v = t[base + j] * s2 + bias[d4 + j];
    if (prev) v += prev[base + j];
    out[base + j] = v;
  }
}

// ---------------- edge scatter: out[col] += T[row] * dis[row]*dis[col] ----------------
// one wave per edge: 32 lanes x float4 = 128 floats
__global__ void k_scatter(const float* __restrict__ t, const float* __restrict__ dis,
                          const long long* __restrict__ ei, float* __restrict__ out, int E) {
  int gid = blockIdx.x * blockDim.x + threadIdx.x;
  int e = gid >> 5;
  int lane = gid & 31;
  if (e >= E) return;
  int r = (int)ei[e];
  int c = (int)ei[(size_t)E + e];
  float norm = dis[r] * dis[c];
  size_t sb = (size_t)r * DH + lane * 4;
  size_t db = (size_t)c * DH + lane * 4;
#pragma unroll
  for (int j = 0; j < 4; ++j)
    atomicAdd(&out[db + j], t[sb + j] * norm);
}

// ---------------- leaky relu in place ----------------
__global__ void k_leaky(float* h, long long n) {
  long long i = (long long)blockIdx.x * blockDim.x + threadIdx.x;
  if (i < n) {
    float v = h[i];
    h[i] = v > 0.0f ? v : NEG_SLOPE * v;
  }
}

// ---------------- head: logits = h @ lw + lb ----------------
__global__ void k_logits(const float* __restrict__ h, const float* __restrict__ lw,
                         const float* __restrict__ lb, float* __restrict__ out, int N) {
  int gid = blockIdx.x * blockDim.x + threadIdx.x;
  int n = gid >> 5;
  int lane = gid & 31;
  if (n >= N) return;
  size_t base = (size_t)n * DH + lane * 4;
  float p = 0.0f;
#pragma unroll
  for (int j = 0; j < 4; ++j) p += h[base + j] * lw[lane * 4 + j];
#pragma unroll
  for (int off = 16; off > 0; off >>= 1) p += __shfl_xor(p, off, 32);
  if (lane == 0) out[n] = p + lb[0];
}

extern "C" void kernel_launch(void* const* d_in, const int* in_sizes, int n_in,
                              void* d_out, int out_size, void* d_ws, size_t ws_size,
                              hipStream_t stream) {
  const float*     x  = (const float*)d_in[0];
  const long long* ei = (const long long*)d_in[1];
  const float* W1 = (const float*)d_in[2]; const float* b1 = (const float*)d_in[3];
  const float* W2 = (const float*)d_in[4]; const float* b2 = (const float*)d_in[5];
  const float* W3 = (const float*)d_in[6]; const float* b3 = (const float*)d_in[7];
  const float* lw = (const float*)d_in[8]; const float* lb = (const float*)d_in[9];
  float* out = (float*)d_out;

  int N = in_sizes[0] / DH;     // 50000
  int E = in_sizes[1] / 2;      // 600000

  // workspace layout: dis | T | A | B   (~77 MB total, L2-resident)
  char* ws = (char*)d_ws;
  float* dis = (float*)ws;
  size_t off = ((size_t)N * 4 + 255) & ~(size_t)255;
  float* T = (float*)(ws + off); off += (size_t)N * DH * 4;
  float* A = (float*)(ws + off); off += (size_t)N * DH * 4;
  float* B = (float*)(ws + off);

  const int TB = 256;
  int gN   = (N + TB - 1) / TB;
  int gE   = (E + TB - 1) / TB;
  int gN32 = (N * 32 + TB - 1) / TB;
  int gE32 = (int)(((long long)E * 32 + TB - 1) / TB);
  long long elems = (long long)N * DH;
  int gElem = (int)((elems + TB - 1) / TB);
  int ntiles = (N + 15) / 16;
  int gGemm = (ntiles + 7) / 8;   // 8 waves per 256-thread block

  // normalization
  k_init_deg <<<gN, TB, 0, stream>>>(dis, N);
  k_count_deg<<<gE, TB, 0, stream>>>(ei, dis, E);
  k_rsqrt    <<<gN, TB, 0, stream>>>(dis, N);

  // layer 1: A = leaky(conv(x, W1, b1))
  k_gemm_wmma<<<gGemm, TB, 0, stream>>>(x, W1, T, N, ntiles);
  k_init_out <<<gN32, TB, 0, stream>>>(A, (const float*)nullptr, T, dis, b1, N);
  k_scatter  <<<gE32, TB, 0, stream>>>(T, dis, ei, A, E);
  k_leaky    <<<gElem, TB, 0, stream>>>(A, elems);

  // layer 2: B = leaky(A + conv(A, W2, b2))
  k_gemm_wmma<<<gGemm, TB, 0, stream>>>(A, W2, T, N, ntiles);
  k_init_out <<<gN32, TB, 0, stream>>>(B, A, T, dis, b2, N);
  k_scatter  <<<gE32, TB, 0, stream>>>(T, dis, ei, B, E);
  k_leaky    <<<gElem, TB, 0, stream>>>(B, elems);

  // layer 3: A = leaky(B + conv(B, W3, b3))
  k_gemm_wmma<<<gGemm, TB, 0, stream>>>(B, W3, T, N, ntiles);
  k_init_out <<<gN32, TB, 0, stream>>>(A, B, T, dis, b3, N);
  k_scatter  <<<gE32, TB, 0, stream>>>(T, dis, ei, A, E);
  k_leaky    <<<gElem, TB, 0, stream>>>(A, elems);

  // head
  k_logits   <<<gN32, TB, 0, stream>>>(A, lw, lb, out, N);
}